// GENEO_MLP_Small_b_13804024889706
// MI455X (gfx1250) — compile-verified
//
#include <hip/hip_runtime.h>
#include <hip/hip_bf16.h>

typedef __attribute__((ext_vector_type(2))) float v2f;
typedef __attribute__((ext_vector_type(8))) float v8f;

#define BATCH 512
#define KCH   64
#define HW    400
#define HPAD  24
#define NCLS  10

// ---------------------------------------------------------------------------
// Kernel 1: per (b,k) compute max and first-argmax of
//   F[b,k,pos] = 1 - (1/25) * sum_{5x5} |pattern[k] - zero-padded-patch|
// One block per batch; 8 waves, each wave owns 8 channels.
// ---------------------------------------------------------------------------
__global__ __launch_bounds__(256) void geneo_max_kernel(
    const float* __restrict__ x,         // (512,1,20,20)
    const float* __restrict__ patterns,  // (64,1,5,5)
    float* __restrict__ maxv,            // (512,64)
    int*   __restrict__ maxi)            // (512,64)
{
    __shared__ float xpad[HPAD * HPAD];   // 24x24 zero-padded image
    __shared__ float pat[KCH * 25];       // all patterns

    const int b   = blockIdx.x;
    const int tid = threadIdx.x;

    for (int i = tid; i < HPAD * HPAD; i += 256) xpad[i] = 0.0f;
    __syncthreads();
    for (int i = tid; i < HW; i += 256) {
        int r = i / 20, c = i % 20;
        xpad[(r + 2) * HPAD + (c + 2)] = x[b * HW + i];
    }
    for (int i = tid; i < KCH * 25; i += 256) pat[i] = patterns[i];
    __syncthreads();

    const int wave = tid >> 5;
    const int lane = tid & 31;

    for (int kk = 0; kk < 8; ++kk) {
        const int k = wave * 8 + kk;
        const float* pk = &pat[k * 25];

        float best = -1e30f;
        int   bidx = 0x7fffffff;
        for (int pos = lane; pos < HW; pos += 32) {
            const int r = pos / 20, c = pos % 20;
            float s = 0.0f;
#pragma unroll
            for (int dr = 0; dr < 5; ++dr)
#pragma unroll
                for (int dc = 0; dc < 5; ++dc)
                    s += fabsf(pk[dr * 5 + dc] - xpad[(r + dr) * HPAD + (c + dc)]);
            const float val = 1.0f - s * 0.04f;   // 1/25
            if (val > best) { best = val; bidx = pos; }  // strict > keeps first index
        }
        // wave32 reduction: max value, smallest index on ties (first-argmax)
#pragma unroll
        for (int off = 16; off >= 1; off >>= 1) {
            const float ov = __shfl_xor(best, off, 32);
            const int   oi = __shfl_xor(bidx, off, 32);
            if (ov > best || (ov == best && oi < bidx)) { best = ov; bidx = oi; }
        }
        if (lane == 0) {
            maxv[b * KCH + k] = best;
            maxi[b * KCH + k] = bidx;
        }
    }
}

// ---------------------------------------------------------------------------
// Kernel 2: scatter sparse contributions -> t = sigmoid(b2 + acc) (16x400),
// then out = sigmoid(t @ wf^T + bf) with V_WMMA_F32_16X16X4_F32.
// One wave (32 threads) per 16-batch tile; 32 blocks total.
// wf is staged into LDS zero-padded to 16 rows so the WMMA B-operand loads
// are branch-free ds_load_b64 (no EXEC-masked loads in the inner loop).
// ---------------------------------------------------------------------------
__global__ __launch_bounds__(32) void finalize_wmma_kernel(
    const float* __restrict__ maxv,   // (512,64)
    const int*   __restrict__ maxi,   // (512,64)
    const float* __restrict__ w2,     // (1,64)
    const float* __restrict__ b2,     // (1,)
    const float* __restrict__ wf,     // (10,400)
    const float* __restrict__ bf,     // (10,)
    float* __restrict__ out)          // (512,10)
{
    __shared__ float t[16 * HW];          // 25.6 KB: A-tile (also accumulator)
    __shared__ float wfs[16 * HW];        // 25.6 KB: wf zero-padded to 16 rows

    const int lane = threadIdx.x;         // 0..31, full wave
    const int b0   = blockIdx.x * 16;

    __builtin_prefetch(wf, 0, 1);         // global_prefetch_b8

    for (int i = lane; i < 16 * HW; i += 32) {
        t[i]   = 0.0f;
        wfs[i] = (i < NCLS * HW) ? wf[i] : 0.0f;  // rows 10..15 zero
    }
    __syncthreads();

    // sparse scatter: 16 batches x 64 channels, one nonzero each
    for (int i = lane; i < 16 * KCH; i += 32) {
        const int bb = i >> 6;
        const int k  = i & 63;
        const float v = maxv[(b0 + bb) * KCH + k] * w2[k];
        const int   p = maxi[(b0 + bb) * KCH + k];
        atomicAdd(&t[bb * HW + p], v);
    }
    __syncthreads();

    const float bias2 = b2[0];
    for (int i = lane; i < 16 * HW; i += 32) {
        const float z = bias2 + t[i];
        t[i] = 1.0f / (1.0f + __expf(-z));
    }
    __syncthreads();

    // fp32 WMMA GEMM: D(16x16) += A(16x4) x B(4x16), 100 K-steps.
    // 32-bit A layout: lanes 0-15 -> M=lane, K={0,1} in the 2 slots;
    //                  lanes 16-31 -> M=lane-16, K={2,3}.
    // B mirrors with N in place of M; B columns 10..15 are zero rows of wfs.
    const int mrow  = lane & 15;
    const int khalf = lane >> 4;
    const float* ta = &t[mrow * HW + khalf * 2];
    const float* tb = &wfs[mrow * HW + khalf * 2];

    v8f acc = {};
#pragma unroll 4
    for (int s = 0; s < 100; ++s) {
        v2f a, bm;
        a.x  = ta[s * 4];
        a.y  = ta[s * 4 + 1];
        bm.x = tb[s * 4];
        bm.y = tb[s * 4 + 1];
        acc = __builtin_amdgcn_wmma_f32_16x16x4_f32(
            /*neg_a=*/false, a, /*neg_b=*/false, bm,
            /*c_mod=*/(short)0, acc, /*reuse_a=*/false, /*reuse_b=*/false);
    }

    // C/D layout: VGPR r -> (M = r + 8*khalf, N = mrow)
    if (mrow < NCLS) {
        const float bb = bf[mrow];
#pragma unroll
        for (int r = 0; r < 8; ++r) {
            const int m = r + 8 * khalf;
            const float z = bb + acc[r];
            out[(b0 + m) * NCLS + mrow] = 1.0f / (1.0f + __expf(-z));
        }
    }
}

// ---------------------------------------------------------------------------
extern "C" void kernel_launch(void* const* d_in, const int* in_sizes, int n_in,
                              void* d_out, int out_size, void* d_ws, size_t ws_size,
                              hipStream_t stream) {
    const float* x        = (const float*)d_in[0];  // (512,1,20,20)
    const float* patterns = (const float*)d_in[1];  // (64,1,5,5)
    const float* w2       = (const float*)d_in[2];  // (1,64)
    const float* b2       = (const float*)d_in[3];  // (1,)
    const float* wf       = (const float*)d_in[4];  // (10,400)
    const float* bf       = (const float*)d_in[5];  // (10,)
    float* out            = (float*)d_out;          // (512,10)

    float* maxv = (float*)d_ws;                       // 512*64 floats
    int*   maxi = (int*)((char*)d_ws + BATCH * KCH * sizeof(float));

    geneo_max_kernel<<<BATCH, 256, 0, stream>>>(x, patterns, maxv, maxi);
    finalize_wmma_kernel<<<BATCH / 16, 32, 0, stream>>>(maxv, maxi, w2, b2, wf, bf, out);
}